// GGNN_80229989089578
// MI455X (gfx1250) — compile-verified
//
#include <hip/hip_runtime.h>
#include <hip/hip_bf16.h>

#define Hdim   128
#define Bdim   512
#define Ndim   40
#define Sdim   50
#define NNODES 100000
#define NOUT   99999          // N_NODES - 1
#define LSTR   132            // padded LDS row stride (dwords): stride%64==4 -> conflict-free frags
#define ASTR   44             // padded adj row stride

typedef __attribute__((ext_vector_type(2))) float v2f;
typedef __attribute__((ext_vector_type(8))) float v8f;

// D = A(16x4 f32) * B(4x16 f32) + C, fp32 WMMA (CDNA5 V_WMMA_F32_16X16X4_F32)
__device__ __forceinline__ v8f wmma4(v2f a, v2f b, v8f c) {
  return __builtin_amdgcn_wmma_f32_16x16x4_f32(false, a, false, b, (short)0, c, false, false);
}

__device__ __forceinline__ float sigmoidf_(float x) { return 1.0f / (1.0f + __expf(-x)); }

// wave index as an SGPR value -> tile loops compile as uniform scalar loops
__device__ __forceinline__ int wave_id() {
  return __builtin_amdgcn_readfirstlane((int)(threadIdx.x >> 5));
}

// ---------------------------------------------------------------- K1: gather
__global__ void ggnn_gather(const int* __restrict__ item,
                            const float* __restrict__ emb,
                            float* __restrict__ fin) {
  int bn = blockIdx.x;            // 0 .. B*N-1
  int t  = threadIdx.x;           // 0 .. 127
  int node = item[bn];
  fin[bn * Hdim + t] = emb[(size_t)node * Hdim + t];
}

// -------------------------------------------- K2: av = [adj_in@(fW_in+b), adj_out@(fW_out+b)]
__global__ void __launch_bounds__(256)
ggnn_av(const float* __restrict__ fin,
        const float* __restrict__ adj_in, const float* __restrict__ adj_out,
        const float* __restrict__ W_in,   const float* __restrict__ b_in,
        const float* __restrict__ W_out,  const float* __restrict__ b_out,
        float* __restrict__ av) {
  __shared__ float sfin[48 * LSTR];          // 48 x 128 (rows 40..47 zero)
  __shared__ float stmp[2][48 * LSTR];       // fin@W + bias
  __shared__ float sadj[2][48 * ASTR];       // adj tiles, rows 40..47 zero

  int b    = blockIdx.x;
  int t    = threadIdx.x;
  int wave = wave_id();
  int lane = t & 31;
  int half = lane >> 4;
  int lid  = lane & 15;

  // zero-padded staging (guards live only in the load phase)
  const float* finb = fin + (size_t)b * Ndim * Hdim;
  for (int i = t; i < 48 * Hdim; i += 256) {
    int r = i >> 7, c = i & 127;
    sfin[r * LSTR + c] = (r < Ndim) ? finb[r * Hdim + c] : 0.f;
  }
  const float* adjb_in  = adj_in  + (size_t)b * Ndim * Ndim;
  const float* adjb_out = adj_out + (size_t)b * Ndim * Ndim;
  for (int i = t; i < 48 * Ndim; i += 256) {
    int r = i / Ndim, k = i % Ndim;
    sadj[0][r * ASTR + k] = (r < Ndim) ? adjb_in[r * Ndim + k]  : 0.f;
    sadj[1][r * ASTR + k] = (r < Ndim) ? adjb_out[r * Ndim + k] : 0.f;
  }
  __syncthreads();

  // phase 1: tmp[io] = fin @ W_io + b_io  (io x2, M tiles x3, N tiles x8 = 48 tiles / 8 waves)
  for (int tile = wave; tile < 48; tile += 8) {       // uniform scalar loop
    int io = tile / 24, rem = tile % 24, mt = rem / 8, ct = rem % 8;
    const float* W    = io ? W_out : W_in;
    const float* bias = io ? b_out : b_in;
    const float* sa   = sfin + (mt * 16 + lid) * LSTR;
    int col = ct * 16 + lid;
    v8f acc = {};
#pragma unroll 4
    for (int k = 0; k < Hdim; k += 4) {
      int kk = k + 2 * half;
      v2f a = *(const v2f*)(sa + kk);
      v2f bf;
      bf.x = W[kk * Hdim + col];
      bf.y = W[(kk + 1) * Hdim + col];
      acc = wmma4(a, bf, acc);
    }
    float bc = bias[col];
    float* st = stmp[io] + mt * 16 * LSTR + col;
#pragma unroll
    for (int i = 0; i < 8; ++i) st[(i + half * 8) * LSTR] = acc[i] + bc;
  }
  __syncthreads();

  // phase 2: a_io = adj_io @ tmp[io] -> av[b][m][io*H + col]
  float* avb = av + (size_t)b * Ndim * 2 * Hdim;
  for (int tile = wave; tile < 48; tile += 8) {       // uniform scalar loop
    int io = tile / 24, rem = tile % 24, mt = rem / 8, ct = rem % 8;
    const float* sa = sadj[io] + (mt * 16 + lid) * ASTR;
    const float* sb = stmp[io];
    int col = ct * 16 + lid;
    v8f acc = {};
#pragma unroll
    for (int k = 0; k < Ndim; k += 4) {      // K = 40
      int kk = k + 2 * half;
      v2f a = *(const v2f*)(sa + kk);
      v2f bf;
      bf.x = sb[kk * LSTR + col];
      bf.y = sb[(kk + 1) * LSTR + col];
      acc = wmma4(a, bf, acc);
    }
#pragma unroll
    for (int i = 0; i < 8; ++i) {
      int m = mt * 16 + i + half * 8;
      if (m < Ndim) avb[m * (2 * Hdim) + io * Hdim + col] = acc[i];
    }
  }
}

// ---------------------------------------------------------------- K3: GRU cell
#define XSTR 388   // 384 + 4 pad
#define GSTR 260   // 256 + 4 pad
__global__ void __launch_bounds__(256)
ggnn_gru(const float* __restrict__ av, const float* __restrict__ fin,
         const float* __restrict__ gate_w, const float* __restrict__ gate_b,
         const float* __restrict__ cand_w, const float* __restrict__ cand_b,
         float* __restrict__ fin2) {
  __shared__ float sx[16 * XSTR];   // x = [av (256) | h (128)], later h <- r*h
  __shared__ float sg[16 * GSTR];   // sigmoid gates [r | z]

  int r0   = blockIdx.x * 16;       // row base in flattened [B*N]
  int t    = threadIdx.x;
  int wave = wave_id();
  int lane = t & 31;
  int half = lane >> 4;
  int lid  = lane & 15;

  for (int i = t; i < 16 * 256; i += 256) {
    int r = i >> 8, c = i & 255;
    sx[r * XSTR + c] = av[(size_t)(r0 + r) * 256 + c];
  }
  for (int i = t; i < 16 * 128; i += 256) {
    int r = i >> 7, c = i & 127;
    sx[r * XSTR + 256 + c] = fin[(size_t)(r0 + r) * 128 + c];
  }
  __syncthreads();

  // gates: [16,384] @ [384,256]
  const float* sa = sx + lid * XSTR;
  for (int ct = wave; ct < 16; ct += 8) {             // uniform scalar loop
    int col = ct * 16 + lid;
    v8f acc = {};
#pragma unroll 4
    for (int k = 0; k < 384; k += 4) {
      int kk = k + 2 * half;
      v2f a = *(const v2f*)(sa + kk);
      v2f bf;
      bf.x = gate_w[kk * 256 + col];
      bf.y = gate_w[(kk + 1) * 256 + col];
      acc = wmma4(a, bf, acc);
    }
    float gb = gate_b[col];
#pragma unroll
    for (int i = 0; i < 8; ++i) {
      int m = i + half * 8;
      sg[m * GSTR + col] = sigmoidf_(acc[i] + gb);
    }
  }
  __syncthreads();

  // x2: last 128 cols become r * h
  for (int i = t; i < 16 * 128; i += 256) {
    int r = i >> 7, c = i & 127;
    sx[r * XSTR + 256 + c] *= sg[r * GSTR + c];
  }
  __syncthreads();

  // candidate: [16,384] @ [384,128], then fused GRU combine
  {
    int col = wave * 16 + lid;
    v8f acc = {};
#pragma unroll 4
    for (int k = 0; k < 384; k += 4) {
      int kk = k + 2 * half;
      v2f a = *(const v2f*)(sa + kk);
      v2f bf;
      bf.x = cand_w[kk * 128 + col];
      bf.y = cand_w[(kk + 1) * 128 + col];
      acc = wmma4(a, bf, acc);
    }
    float cb = cand_b[col];
#pragma unroll
    for (int i = 0; i < 8; ++i) {
      int m = i + half * 8;
      float c = tanhf(acc[i] + cb);
      float z = sg[m * GSTR + 128 + col];
      float h = fin[(size_t)(r0 + m) * 128 + col];
      fin2[(size_t)(r0 + m) * 128 + col] = z * h + (1.f - z) * c;
    }
  }
}

// ---------------------------------------------------------------- K4: attention readout
__global__ void __launch_bounds__(128)
ggnn_attn(const int* __restrict__ alias, const float* __restrict__ mask,
          const float* __restrict__ fin2,
          const float* __restrict__ w1, const float* __restrict__ w2,
          const float* __restrict__ v,  const float* __restrict__ nasr_b,
          float* __restrict__ ma) {
  __shared__ float sseq[Sdim][Hdim];
  __shared__ float slh[Hdim];
  __shared__ float red[Hdim];

  int b = blockIdx.x;
  int j = threadIdx.x;                     // 0..127

  int rm = 0;
  for (int s = 0; s < Sdim; ++s) rm += (int)mask[b * Sdim + s];
  int last_id = alias[b * Sdim + (rm - 1)];

  const float* finb = fin2 + (size_t)b * Ndim * Hdim;
  for (int s = 0; s < Sdim; ++s) {
    int a = alias[b * Sdim + s];
    sseq[s][j] = finb[a * Hdim + j];
  }
  slh[j] = finb[last_id * Hdim + j];
  __syncthreads();

  float lastj = 0.f;
#pragma unroll 8
  for (int k = 0; k < Hdim; ++k) lastj += slh[k] * w1[k * Hdim + j];
  float vj = v[j];
  float bj = nasr_b[j];
  float maj = 0.f;

  for (int s = 0; s < Sdim; ++s) {
    float seq = 0.f;
#pragma unroll 8
    for (int k = 0; k < Hdim; ++k) seq += sseq[s][k] * w2[k * Hdim + j];
    float m = sigmoidf_(lastj + seq + bj);
    red[j] = m * vj;
    __syncthreads();
    for (int off = 64; off > 0; off >>= 1) {
      if (j < off) red[j] += red[j + off];
      __syncthreads();
    }
    float coef = red[0] * mask[b * Sdim + s];
    __syncthreads();
    maj += coef * sseq[s][j];
  }
  ma[b * Hdim + j] = maj;
}

// ---------------------------------------------------------------- K5: logits = ma @ emb[1:]^T
__global__ void __launch_bounds__(256)
ggnn_logits(const float* __restrict__ ma, const float* __restrict__ emb,
            float* __restrict__ out) {
  __shared__ float sma[16 * LSTR];
  __shared__ float semb[128 * LSTR];

  int j0   = blockIdx.x * 128;   // node-column base
  int m0   = blockIdx.y * 16;    // batch-row base
  int t    = threadIdx.x;
  int wave = wave_id();
  int lane = t & 31;
  int half = lane >> 4;
  int lid  = lane & 15;

  for (int i = t; i < 16 * 128; i += 256) {
    int r = i >> 7, c = i & 127;
    sma[r * LSTR + c] = ma[(m0 + r) * Hdim + c];
  }
  for (int i = t; i < 128 * 128; i += 256) {
    int r = i >> 7, c = i & 127;
    int col = j0 + r;
    semb[r * LSTR + c] = (col < NOUT) ? emb[(size_t)(col + 1) * Hdim + c] : 0.f;
  }
  // hint the next emb tile toward L2 (gfx1250 global_prefetch path)
  if (t < 128 && (j0 + 128 + t) < NOUT)
    __builtin_prefetch(emb + (size_t)(j0 + 129 + t) * Hdim, 0, 1);
  __syncthreads();

  int n0 = wave * 16;
  const float* sa = sma + lid * LSTR;
  const float* sb = semb + (n0 + lid) * LSTR;
  v8f acc = {};
#pragma unroll 8
  for (int k = 0; k < Hdim; k += 4) {
    int kk = k + 2 * half;
    v2f a  = *(const v2f*)(sa + kk);
    v2f bf = *(const v2f*)(sb + kk);
    acc = wmma4(a, bf, acc);
  }
#pragma unroll
  for (int i = 0; i < 8; ++i) {
    int m   = m0 + i + half * 8;
    int col = j0 + n0 + lid;
    if (col < NOUT) out[(size_t)m * NOUT + col] = acc[i];
  }
}

// ---------------------------------------------------------------- launcher
extern "C" void kernel_launch(void* const* d_in, const int* in_sizes, int n_in,
                              void* d_out, int out_size, void* d_ws, size_t ws_size,
                              hipStream_t stream) {
  const int*   item    = (const int*)d_in[0];
  const int*   alias   = (const int*)d_in[1];
  const float* mask    = (const float*)d_in[2];
  const float* adj_in  = (const float*)d_in[3];
  const float* adj_out = (const float*)d_in[4];
  const float* emb     = (const float*)d_in[5];
  const float* W_in    = (const float*)d_in[6];
  const float* b_in    = (const float*)d_in[7];
  const float* W_out   = (const float*)d_in[8];
  const float* b_out   = (const float*)d_in[9];
  const float* gate_w  = (const float*)d_in[10];
  const float* gate_b  = (const float*)d_in[11];
  const float* cand_w  = (const float*)d_in[12];
  const float* cand_b  = (const float*)d_in[13];
  const float* w1      = (const float*)d_in[14];
  const float* w2      = (const float*)d_in[15];
  const float* nasr_v  = (const float*)d_in[16];
  const float* nasr_b  = (const float*)d_in[17];
  float* out = (float*)d_out;

  // workspace layout (floats)
  float* fin  = (float*)d_ws;                        // B*N*H
  float* av   = fin  + (size_t)Bdim * Ndim * Hdim;   // B*N*2H
  float* fin2 = av   + (size_t)Bdim * Ndim * 2 * Hdim;
  float* mabf = fin2 + (size_t)Bdim * Ndim * Hdim;   // B*H

  ggnn_gather<<<Bdim * Ndim, 128, 0, stream>>>(item, emb, fin);
  ggnn_av<<<Bdim, 256, 0, stream>>>(fin, adj_in, adj_out, W_in, b_in, W_out, b_out, av);
  ggnn_gru<<<(Bdim * Ndim) / 16, 256, 0, stream>>>(av, fin, gate_w, gate_b, cand_w, cand_b, fin2);
  ggnn_attn<<<Bdim, 128, 0, stream>>>(alias, mask, fin2, w1, w2, nasr_v, nasr_b, mabf);
  ggnn_logits<<<dim3((NOUT + 127) / 128, Bdim / 16), 256, 0, stream>>>(mabf, emb, out);
}